// MoELayer_24541443129819
// MI455X (gfx1250) — compile-verified
//
#include <hip/hip_runtime.h>
#include <math.h>

#define NTOK 2048      // B*L
#define D_   768
#define E_   16
#define F_   3072
#define SLOTS 4096     // NTOK * K(=2)
#define PADCAP 4608    // SLOTS + 16 experts * 31 pad rows, rounded to 32
#define MAXTILES 144   // sum ceil(cnt_e/32) <= 4096/32 + 16

typedef float v2f __attribute__((ext_vector_type(2)));
typedef float v8f __attribute__((ext_vector_type(8)));

__device__ __forceinline__ float gelu_erf(float v) {
  return 0.5f * v * (1.0f + erff(v * 0.70710678118654752f));
}

__device__ __forceinline__ v8f wmma4(v2f a, v2f b, v8f c) {
  return __builtin_amdgcn_wmma_f32_16x16x4_f32(false, a, false, b, (short)0, c, false, false);
}

// CDNA5 async copy: per-lane 16B global -> LDS, tracked by ASYNCcnt
__device__ __forceinline__ void async_b128(unsigned lds_off, const void* gaddr) {
  asm volatile("global_load_async_to_lds_b128 %0, %1, off"
               :: "v"(lds_off), "v"(gaddr) : "memory");
}
__device__ __forceinline__ void wait_async_20() {
  asm volatile("s_wait_asynccnt 20" ::: "memory");
}
__device__ __forceinline__ void wait_async_0() {
  asm volatile("s_wait_asynccnt 0" ::: "memory");
}

// ---------------- router: logits, softmax, top-2 ----------------
__global__ void k_router(const float* __restrict__ x, const float* __restrict__ wr,
                         int* __restrict__ eid, float* __restrict__ gv) {
  int wave = threadIdx.x >> 5;
  int lane = threadIdx.x & 31;
  int t = blockIdx.x * 8 + wave;
  if (t >= NTOK) return;
  float acc[E_];
#pragma unroll
  for (int e = 0; e < E_; ++e) acc[e] = 0.f;
  const float* xr = x + (size_t)t * D_;
  for (int d = lane; d < D_; d += 32) {
    float xv = xr[d];
#pragma unroll
    for (int e = 0; e < E_; ++e) acc[e] += xv * wr[e * D_ + d];
  }
#pragma unroll
  for (int e = 0; e < E_; ++e) {
    float v = acc[e];
#pragma unroll
    for (int m = 16; m >= 1; m >>= 1) v += __shfl_xor(v, m, 32);
    acc[e] = v;
  }
  if (lane == 0) {
    float mx = acc[0];
    for (int e = 1; e < E_; ++e) mx = fmaxf(mx, acc[e]);
    float sum = 0.f;
    for (int e = 0; e < E_; ++e) sum += expf(acc[e] - mx);
    int e0 = 0; float v0 = acc[0];
    for (int e = 1; e < E_; ++e) if (acc[e] > v0) { v0 = acc[e]; e0 = e; }
    int e1 = (e0 == 0) ? 1 : 0; float v1 = acc[e1];
    for (int e = 0; e < E_; ++e) if (e != e0 && acc[e] > v1) { v1 = acc[e]; e1 = e; }
    float inv = 1.f / sum;
    eid[t * 2 + 0] = e0; gv[t * 2 + 0] = expf(v0 - mx) * inv;
    eid[t * 2 + 1] = e1; gv[t * 2 + 1] = expf(v1 - mx) * inv;
  }
}

// ---------------- bookkeeping ----------------
__global__ void k_zero(int* __restrict__ counts, int* __restrict__ perm) {
  int i = threadIdx.x;
  if (i < E_) counts[i] = 0;
  for (int p = i; p < PADCAP; p += 256) perm[p] = -1;
}

__global__ void k_count(const int* __restrict__ eid, int* __restrict__ counts) {
  int s = blockIdx.x * 256 + threadIdx.x;
  if (s < SLOTS) atomicAdd(&counts[eid[s]], 1);
}

__global__ void k_offsets(const int* __restrict__ counts, int* __restrict__ offsets,
                          int* __restrict__ ntile, int* __restrict__ tileE,
                          int* __restrict__ tileRow) {
  if (threadIdx.x != 0) return;
  int base = 0, t = 0;
  for (int e = 0; e < E_; ++e) {
    offsets[e] = base;
    int nt = (counts[e] + 31) >> 5;
    for (int i = 0; i < nt; ++i) { tileE[t] = e; tileRow[t] = base + 32 * i; ++t; }
    base += nt << 5;
  }
  ntile[0] = t;
}

// deterministic per-expert compaction: one wave per expert, ballot prefix
__global__ void k_build(const int* __restrict__ eid, const int* __restrict__ offsets,
                        int* __restrict__ perm) {
  int e = blockIdx.x;
  int lane = threadIdx.x;
  int cur = offsets[e];
  for (int s0 = 0; s0 < SLOTS; s0 += 32) {
    int s = s0 + lane;
    bool m = (eid[s] == e);
    unsigned mask = __builtin_amdgcn_ballot_w32(m);
    if (m) {
      int below = __builtin_popcount(mask & ((1u << lane) - 1u));
      perm[cur + below] = s;
    }
    cur += __builtin_popcount(mask);
  }
}

// ---------------- fc1: H = gelu(X @ w1[e]^T + b1[e]) ----------------
// block tile: 32 rows x 128 F-cols; 4 waves, 2x2 WMMA tiles each;
// double-buffered LDS fed by async global->LDS copies.
__global__ __launch_bounds__(128) void k_fc1(
    const float* __restrict__ x, const float* __restrict__ w1, const float* __restrict__ b1,
    const int* __restrict__ perm, const int* __restrict__ ntile,
    const int* __restrict__ tileE, const int* __restrict__ tileRow,
    float* __restrict__ H) {
  int tile = blockIdx.y;
  if (tile >= ntile[0]) return;
  int e = tileE[tile];
  int rowbase = tileRow[tile];
  int f0 = blockIdx.x * 128;
  __shared__ float Xs[2][32 * 64];    // 2 x  8 KB
  __shared__ float Ws[2][128 * 64];   // 2 x 32 KB
  __shared__ int toks[32];
  int tid = threadIdx.x, wave = tid >> 5, lane = tid & 31;
  if (tid < 32) toks[tid] = perm[rowbase + tid];
  __syncthreads();

  int r = tid >> 2, cq = (tid & 3) * 16;
  int sa = toks[r];
  int tok = (sa >= 0) ? (sa >> 1) : 0;   // clamp pad rows to a valid address
  const float* gx = x + (size_t)tok * D_ + cq;
  const float* gw = w1 + ((size_t)e * F_ + f0 + tid) * D_;
  unsigned xb = (unsigned)(size_t)&Xs[0][0] + (unsigned)((r * 64 + cq) * 4);
  unsigned wb = (unsigned)(size_t)&Ws[0][0] + (unsigned)(tid * 64 * 4);

#pragma unroll
  for (int j = 0; j < 4; ++j)  async_b128(xb + j * 16, gx + j * 4);
#pragma unroll
  for (int j = 0; j < 16; ++j) async_b128(wb + j * 16, gw + j * 4);

  v8f c00 = {0.f,0.f,0.f,0.f,0.f,0.f,0.f,0.f};
  v8f c01 = c00, c10 = c00, c11 = c00;
  int n = lane & 15, hi = lane >> 4;

  for (int i = 0; i < 12; ++i) {
    int buf = i & 1;
    if (i + 1 < 12) {
      int kc = (i + 1) * 64;
      unsigned xo = xb + (unsigned)((buf ^ 1) * (32 * 64 * 4));
      unsigned wo = wb + (unsigned)((buf ^ 1) * (128 * 64 * 4));
#pragma unroll
      for (int j = 0; j < 4; ++j)  async_b128(xo + j * 16, gx + kc + j * 4);
#pragma unroll
      for (int j = 0; j < 16; ++j) async_b128(wo + j * 16, gw + kc + j * 4);
      wait_async_20();
    } else {
      wait_async_0();
    }
    __syncthreads();
    const float* Xb = &Xs[buf][0];
    const float* Wb = &Ws[buf][0];
    const float* ax0 = &Xb[n * 64 + 2 * hi];
    const float* ax1 = &Xb[(16 + n) * 64 + 2 * hi];
    const float* bx0 = &Wb[(wave * 32 + n) * 64 + 2 * hi];
    const float* bx1 = &Wb[(wave * 32 + 16 + n) * 64 + 2 * hi];
#pragma unroll
    for (int kk = 0; kk < 16; ++kk) {
      v2f a0 = *(const v2f*)(ax0 + kk * 4);
      v2f a1 = *(const v2f*)(ax1 + kk * 4);
      v2f b0 = *(const v2f*)(bx0 + kk * 4);
      v2f b1v = *(const v2f*)(bx1 + kk * 4);
      c00 = wmma4(a0, b0, c00);
      c01 = wmma4(a0, b1v, c01);
      c10 = wmma4(a1, b0, c10);
      c11 = wmma4(a1, b1v, c11);
    }
    __syncthreads();
  }

  int fA = f0 + wave * 32 + n;
  int fB = fA + 16;
  float biasA = b1[e * F_ + fA];
  float biasB = b1[e * F_ + fB];
#pragma unroll
  for (int v = 0; v < 8; ++v) {
    int m0 = v + 8 * hi;
    int m1 = 16 + m0;
    H[(size_t)(rowbase + m0) * F_ + fA] = gelu_erf(c00[v] + biasA);
    H[(size_t)(rowbase + m0) * F_ + fB] = gelu_erf(c01[v] + biasB);
    H[(size_t)(rowbase + m1) * F_ + fA] = gelu_erf(c10[v] + biasA);
    H[(size_t)(rowbase + m1) * F_ + fB] = gelu_erf(c11[v] + biasB);
  }
}

// ---------------- fc2: contrib[slot] = gate * (H @ w2[e]^T + b2[e]) ----------------
__global__ __launch_bounds__(128) void k_fc2(
    const float* __restrict__ H, const float* __restrict__ w2, const float* __restrict__ b2,
    const int* __restrict__ perm, const float* __restrict__ gv,
    const int* __restrict__ ntile, const int* __restrict__ tileE,
    const int* __restrict__ tileRow, float* __restrict__ contrib) {
  int tile = blockIdx.y;
  if (tile >= ntile[0]) return;
  int e = tileE[tile];
  int rowbase = tileRow[tile];
  int d0 = blockIdx.x * 128;
  __shared__ float Hs[2][32 * 64];
  __shared__ float Ws[2][128 * 64];
  __shared__ int toks[32];
  int tid = threadIdx.x, wave = tid >> 5, lane = tid & 31;
  if (tid < 32) toks[tid] = perm[rowbase + tid];
  __syncthreads();

  int r = tid >> 2, cq = (tid & 3) * 16;
  const float* gh = H + (size_t)(rowbase + r) * F_ + cq;
  const float* gw = w2 + ((size_t)e * D_ + d0 + tid) * F_;
  unsigned hb = (unsigned)(size_t)&Hs[0][0] + (unsigned)((r * 64 + cq) * 4);
  unsigned wb = (unsigned)(size_t)&Ws[0][0] + (unsigned)(tid * 64 * 4);

#pragma unroll
  for (int j = 0; j < 4; ++j)  async_b128(hb + j * 16, gh + j * 4);
#pragma unroll
  for (int j = 0; j < 16; ++j) async_b128(wb + j * 16, gw + j * 4);

  v8f c00 = {0.f,0.f,0.f,0.f,0.f,0.f,0.f,0.f};
  v8f c01 = c00, c10 = c00, c11 = c00;
  int n = lane & 15, hi = lane >> 4;

  for (int i = 0; i < 48; ++i) {
    int buf = i & 1;
    if (i + 1 < 48) {
      int kc = (i + 1) * 64;
      unsigned ho = hb + (unsigned)((buf ^ 1) * (32 * 64 * 4));
      unsigned wo = wb + (unsigned)((buf ^ 1) * (128 * 64 * 4));
#pragma unroll
      for (int j = 0; j < 4; ++j)  async_b128(ho + j * 16, gh + kc + j * 4);
#pragma unroll
      for (int j = 0; j < 16; ++j) async_b128(wo + j * 16, gw + kc + j * 4);
      wait_async_20();
    } else {
      wait_async_0();
    }
    __syncthreads();
    const float* Hb = &Hs[buf][0];
    const float* Wb = &Ws[buf][0];
    const float* ax0 = &Hb[n * 64 + 2 * hi];
    const float* ax1 = &Hb[(16 + n) * 64 + 2 * hi];
    const float* bx0 = &Wb[(wave * 32 + n) * 64 + 2 * hi];
    const float* bx1 = &Wb[(wave * 32 + 16 + n) * 64 + 2 * hi];
#pragma unroll
    for (int kk = 0; kk < 16; ++kk) {
      v2f a0 = *(const v2f*)(ax0 + kk * 4);
      v2f a1 = *(const v2f*)(ax1 + kk * 4);
      v2f b0 = *(const v2f*)(bx0 + kk * 4);
      v2f b1v = *(const v2f*)(bx1 + kk * 4);
      c00 = wmma4(a0, b0, c00);
      c01 = wmma4(a0, b1v, c01);
      c10 = wmma4(a1, b0, c10);
      c11 = wmma4(a1, b1v, c11);
    }
    __syncthreads();
  }

  int dA = d0 + wave * 32 + n;
  int dB = dA + 16;
  float biasA = b2[e * D_ + dA];
  float biasB = b2[e * D_ + dB];
#pragma unroll
  for (int v = 0; v < 8; ++v) {
    int m0 = v + 8 * hi;
    int m1 = 16 + m0;
    int s0 = toks[m0];
    int s1 = toks[m1];
    if (s0 >= 0) {
      float g = gv[s0];
      contrib[(size_t)s0 * D_ + dA] = (c00[v] + biasA) * g;
      contrib[(size_t)s0 * D_ + dB] = (c01[v] + biasB) * g;
    }
    if (s1 >= 0) {
      float g = gv[s1];
      contrib[(size_t)s1 * D_ + dA] = (c10[v] + biasA) * g;
      contrib[(size_t)s1 * D_ + dB] = (c11[v] + biasB) * g;
    }
  }
}

// ---------------- combine: out = x + c(top1) + c(top2) ----------------
__global__ void k_combine(const float* __restrict__ x, const float* __restrict__ contrib,
                          float* __restrict__ out) {
  int t = blockIdx.y;
  int d = blockIdx.x * 256 + threadIdx.x;
  size_t i = (size_t)t * D_ + d;
  out[i] = x[i] + contrib[(size_t)(2 * t) * D_ + d] + contrib[(size_t)(2 * t + 1) * D_ + d];
}

extern "C" void kernel_launch(void* const* d_in, const int* in_sizes, int n_in,
                              void* d_out, int out_size, void* d_ws, size_t ws_size,
                              hipStream_t stream) {
  (void)in_sizes; (void)n_in; (void)out_size; (void)ws_size;
  const float* x  = (const float*)d_in[0];
  const float* wr = (const float*)d_in[1];
  const float* w1 = (const float*)d_in[2];
  const float* b1 = (const float*)d_in[3];
  const float* w2 = (const float*)d_in[4];
  const float* b2 = (const float*)d_in[5];
  float* out = (float*)d_out;

  // workspace layout (ints then floats; H aligned to 16B)
  int* counts  = (int*)d_ws;           // 16
  int* offsets = counts + 16;          // 16
  int* ntile   = offsets + 16;         // 1 (+1 pad)
  int* tileE   = ntile + 2;            // 144
  int* tileRow = tileE + MAXTILES;     // 144
  int* eid     = tileRow + MAXTILES;   // 4096
  int* perm    = eid + SLOTS;          // 4608
  float* gv    = (float*)(perm + PADCAP);  // 4096
  size_t ofs = (size_t)((gv + SLOTS) - (float*)d_ws);
  ofs = (ofs + 3) & ~(size_t)3;        // 16-byte align
  float* H = (float*)d_ws + ofs;                     // PADCAP * F_
  float* contrib = H + (size_t)PADCAP * F_;          // SLOTS * D_

  k_router <<<NTOK / 8, 256, 0, stream>>>(x, wr, eid, gv);
  k_zero   <<<1, 256, 0, stream>>>(counts, perm);
  k_count  <<<SLOTS / 256, 256, 0, stream>>>(eid, counts);
  k_offsets<<<1, 32, 0, stream>>>(counts, offsets, ntile, tileE, tileRow);
  k_build  <<<E_, 32, 0, stream>>>(eid, offsets, perm);
  k_fc1    <<<dim3(F_ / 128, MAXTILES), 128, 0, stream>>>(x, w1, b1, perm, ntile, tileE, tileRow, H);
  k_fc2    <<<dim3(D_ / 128, MAXTILES), 128, 0, stream>>>(H, w2, b2, perm, gv, ntile, tileE, tileRow, contrib);
  k_combine<<<dim3(3, NTOK), 256, 0, stream>>>(x, contrib, out);
}